// StrangeAttractorPooling_71683004170584
// MI455X (gfx1250) — compile-verified
//
#include <hip/hip_runtime.h>
#include <hip/hip_bf16.h>

typedef __attribute__((ext_vector_type(2))) float v2f;
typedef __attribute__((ext_vector_type(4))) float v4f;
typedef __attribute__((ext_vector_type(8))) float v8f;

#define BDIM 32
#define SDIM 4096
#define DDIM 512
#define NCHUNK 8
#define WAVES_PER_WG 8
#define TOKENS_PER_WAVE (SDIM / NCHUNK / WAVES_PER_WG) /* 64 */
#define NPART (NCHUNK * WAVES_PER_WG)                  /* 64 partial slabs per batch */
#define KD 2048                                        /* K*D */
#define OUTD 512

// ---------------------------------------------------------------------------
// Kernel 1: fused streaming pass. One wave owns TOKENS_PER_WAVE tokens of one
// batch. x is read exactly once (non-temporal). Per-token attractor weights
// are computed with a wave32 shfl_xor reduction; weighted x accumulates in
// 64 registers/lane; wave writes a private partial slab (deterministic).
// ---------------------------------------------------------------------------
__global__ __launch_bounds__(256) void pool_stream_kernel(
    const float* __restrict__ x, const float* __restrict__ centers,
    const float* __restrict__ Wa, const float* __restrict__ ba,
    const float* __restrict__ Wh, const float* __restrict__ bh,
    const float* __restrict__ Wk, const float* __restrict__ bk,
    float* __restrict__ P)
{
  const int lane  = threadIdx.x & 31;
  const int wave  = threadIdx.x >> 5;
  const int b     = blockIdx.x;
  const int chunk = blockIdx.y;

  // Per-lane slice of Wa [512,3]: this lane's 16 d-indices, 3 cols each.
  float wa[48];
#pragma unroll
  for (int i = 0; i < 4; ++i)
#pragma unroll
    for (int j = 0; j < 4; ++j) {
      const int d = i * 128 + lane * 4 + j;
#pragma unroll
      for (int a = 0; a < 3; ++a) wa[(i * 4 + j) * 3 + a] = Wa[d * 3 + a];
    }

  // Uniform small params -> scalar loads -> SGPRs.
  float Cc[12], WhR[18], bhR[6], WkR[24], bkR[4], baR[3];
#pragma unroll
  for (int i = 0; i < 12; ++i) Cc[i] = centers[i];
#pragma unroll
  for (int i = 0; i < 18; ++i) WhR[i] = Wh[i];
#pragma unroll
  for (int i = 0; i < 6; ++i) bhR[i] = bh[i];
#pragma unroll
  for (int i = 0; i < 24; ++i) WkR[i] = Wk[i];
#pragma unroll
  for (int i = 0; i < 4; ++i) bkR[i] = bk[i];
#pragma unroll
  for (int i = 0; i < 3; ++i) baR[i] = ba[i];

  float acc[64];
#pragma unroll
  for (int i = 0; i < 64; ++i) acc[i] = 0.0f;

  const int s0 = chunk * (SDIM / NCHUNK) + wave * TOKENS_PER_WAVE;
  const float* xb = x + ((size_t)b * SDIM + s0) * DDIM;

  for (int t = 0; t < TOKENS_PER_WAVE; ++t) {
    const float* row = xb + (size_t)t * DDIM;
    v4f xv[4];
#pragma unroll
    for (int i = 0; i < 4; ++i)
      xv[i] = __builtin_nontemporal_load((const v4f*)(row + i * 128 + lane * 4));

    // Partial dot-products for the 3 attractor projections.
    float p0 = 0.f, p1 = 0.f, p2 = 0.f;
#pragma unroll
    for (int i = 0; i < 4; ++i)
#pragma unroll
      for (int j = 0; j < 4; ++j) {
        const float v = xv[i][j];
        const int w = (i * 4 + j) * 3;
        p0 = fmaf(v, wa[w + 0], p0);
        p1 = fmaf(v, wa[w + 1], p1);
        p2 = fmaf(v, wa[w + 2], p2);
      }
    // wave32 all-reduce
#pragma unroll
    for (int off = 16; off >= 1; off >>= 1) {
      p0 += __shfl_xor(p0, off, 32);
      p1 += __shfl_xor(p1, off, 32);
      p2 += __shfl_xor(p2, off, 32);
    }
    const float af0 = p0 + baR[0];
    const float af1 = p1 + baR[1];
    const float af2 = p2 + baR[2];

    // L2 distance to each attractor center.
    float dist[4];
#pragma unroll
    for (int k = 0; k < 4; ++k) {
      const float d0 = af0 - Cc[k * 3 + 0];
      const float d1 = af1 - Cc[k * 3 + 1];
      const float d2 = af2 - Cc[k * 3 + 2];
      dist[k] = sqrtf(d0 * d0 + d1 * d1 + d2 * d2);
    }
    // MLP: relu(af @ Wh + bh) @ Wk + bk
    float h[6];
#pragma unroll
    for (int j = 0; j < 6; ++j) {
      const float v =
          fmaf(af0, WhR[j], fmaf(af1, WhR[6 + j], fmaf(af2, WhR[12 + j], bhR[j])));
      h[j] = v > 0.f ? v : 0.f;
    }
    float basek[4];
#pragma unroll
    for (int k = 0; k < 4; ++k) {
      float v = bkR[k];
#pragma unroll
      for (int j = 0; j < 6; ++j) v = fmaf(h[j], WkR[j * 4 + k], v);
      basek[k] = v;
    }
    // softmax(-dist)
    float m1 = -dist[0];
#pragma unroll
    for (int k = 1; k < 4; ++k) m1 = fmaxf(m1, -dist[k]);
    float e[4], s1 = 0.f;
#pragma unroll
    for (int k = 0; k < 4; ++k) { e[k] = __expf(-dist[k] - m1); s1 += e[k]; }
    const float inv1 = 1.0f / s1;
    // softmax(base + distance_weights)
    float z[4];
#pragma unroll
    for (int k = 0; k < 4; ++k) z[k] = basek[k] + e[k] * inv1;
    float m2 = z[0];
#pragma unroll
    for (int k = 1; k < 4; ++k) m2 = fmaxf(m2, z[k]);
    float f[4], s2 = 0.f;
#pragma unroll
    for (int k = 0; k < 4; ++k) { f[k] = __expf(z[k] - m2); s2 += f[k]; }
    const float inv2 = 1.0f / s2;

    // Weighted accumulate into per-lane registers.
#pragma unroll
    for (int k = 0; k < 4; ++k) {
      const float w = f[k] * inv2;
#pragma unroll
      for (int i = 0; i < 4; ++i)
#pragma unroll
        for (int j = 0; j < 4; ++j)
          acc[k * 16 + i * 4 + j] = fmaf(w, xv[i][j], acc[k * 16 + i * 4 + j]);
    }
  }

  // Private partial slab: P[b][chunk][wave][k*512 + d]
  float* dst = P + (((size_t)(b * NCHUNK + chunk) * WAVES_PER_WG) + wave) * KD;
#pragma unroll
  for (int k = 0; k < 4; ++k)
#pragma unroll
    for (int i = 0; i < 4; ++i) {
      v4f vv;
#pragma unroll
      for (int j = 0; j < 4; ++j) vv[j] = acc[k * 16 + i * 4 + j];
      *(v4f*)(dst + k * 512 + i * 128 + lane * 4) = vv;
    }
}

// ---------------------------------------------------------------------------
// Kernel 2: deterministic fixed-order reduction of the 64 partial slabs.
// ---------------------------------------------------------------------------
__global__ __launch_bounds__(256) void reduce_partials(
    const float* __restrict__ P, float* __restrict__ pooled)
{
  const int idx = blockIdx.x * 256 + threadIdx.x;  // 0 .. 32*2048-1
  const int b = idx >> 11;
  const int e = idx & (KD - 1);
  float s = 0.f;
  for (int c = 0; c < NPART; ++c)
    s += P[((size_t)b * NPART + c) * KD + e];
  pooled[idx] = s;
}

// ---------------------------------------------------------------------------
// Kernel 3: pooled[32,2048] @ Wout[2048,512] + bout via V_WMMA_F32_16X16X4_F32.
// One wave per 16x16 output tile (2 M-tiles x 32 N-tiles = 64 waves).
// A 16x4 layout: lanes 0-15 hold M=lane, VGPR0/1 = K0/K1; lanes 16-31 = K2/K3.
// B 4x16: VGPR0 lanes 0-15 = row K0, lanes 16-31 = row K2; VGPR1 = K1/K3.
// ---------------------------------------------------------------------------
__global__ __launch_bounds__(256) void wmma_out_gemm(
    const float* __restrict__ pooled, const float* __restrict__ Wout,
    const float* __restrict__ bout, float* __restrict__ out)
{
  const int lane = threadIdx.x & 31;
  const int wave = threadIdx.x >> 5;
  const int task = blockIdx.x * WAVES_PER_WG + wave;  // 0..63
  const int m0 = (task >> 5) * 16;                    // 0 or 16
  const int n0 = (task & 31) * 16;
  const int mrow = m0 + (lane & 15);
  const int koff = (lane >> 4) << 1;  // 0 for lanes 0-15, 2 for 16-31
  const int ncol = n0 + (lane & 15);

  v8f c0 = {}; v8f c1 = {};
  for (int k0 = 0; k0 < KD; k0 += 8) {
    {
      v2f a = *(const v2f*)(pooled + (size_t)mrow * KD + k0 + koff);
      v2f bb;
      bb[0] = Wout[(size_t)(k0 + koff) * OUTD + ncol];
      bb[1] = Wout[(size_t)(k0 + koff + 1) * OUTD + ncol];
      c0 = __builtin_amdgcn_wmma_f32_16x16x4_f32(false, a, false, bb,
                                                 (short)0, c0, false, false);
    }
    {
      v2f a = *(const v2f*)(pooled + (size_t)mrow * KD + k0 + 4 + koff);
      v2f bb;
      bb[0] = Wout[(size_t)(k0 + 4 + koff) * OUTD + ncol];
      bb[1] = Wout[(size_t)(k0 + 4 + koff + 1) * OUTD + ncol];
      c1 = __builtin_amdgcn_wmma_f32_16x16x4_f32(false, a, false, bb,
                                                 (short)0, c1, false, false);
    }
  }
  const v8f c = c0 + c1;

  // C/D layout: VGPR r, lanes 0-15 -> M=r; lanes 16-31 -> M=8+r; N=lane&15.
  const int mbase = m0 + ((lane >> 4) << 3);
  const float bo = bout[ncol];
#pragma unroll
  for (int r = 0; r < 8; ++r)
    out[(size_t)(mbase + r) * OUTD + ncol] = c[r] + bo;
}

// ---------------------------------------------------------------------------
extern "C" void kernel_launch(void* const* d_in, const int* in_sizes, int n_in,
                              void* d_out, int out_size, void* d_ws, size_t ws_size,
                              hipStream_t stream) {
  const float* x       = (const float*)d_in[0];
  const float* centers = (const float*)d_in[1];
  const float* Wa      = (const float*)d_in[2];
  const float* ba      = (const float*)d_in[3];
  const float* Wh      = (const float*)d_in[4];
  const float* bh      = (const float*)d_in[5];
  const float* Wk      = (const float*)d_in[6];
  const float* bk      = (const float*)d_in[7];
  const float* Wout    = (const float*)d_in[8];
  const float* bout    = (const float*)d_in[9];
  float* out = (float*)d_out;

  float* P      = (float*)d_ws;                       // 32*64*2048 fp32 = 16 MB
  float* pooled = P + (size_t)BDIM * NPART * KD;      // 32*2048 fp32

  dim3 g1(BDIM, NCHUNK);
  pool_stream_kernel<<<g1, 256, 0, stream>>>(x, centers, Wa, ba, Wh, bh, Wk, bk, P);
  reduce_partials<<<(BDIM * KD) / 256, 256, 0, stream>>>(P, pooled);
  wmma_out_gemm<<<(64 / WAVES_PER_WG), 256, 0, stream>>>(pooled, Wout, bout, out);
}